// Embedding_ready_75453985456999
// MI455X (gfx1250) — compile-verified
//
#include <hip/hip_runtime.h>
#include <hip/hip_bf16.h>
#include <stdint.h>

typedef __attribute__((ext_vector_type(4))) float v4f;

#define B    256
#define S    200
#define D    1024
#define H    16
#define PAD  12975
#define TOK  8          // tokens per block (8 divides 200 -> blocks never cross a batch row)

// ---- CDNA5 async global->LDS copy helpers (ASYNCcnt-tracked) --------------
__device__ __forceinline__ void async_row_load16(unsigned ldsOff, const v4f* gsrc)
{
    asm volatile("global_load_async_to_lds_b128 %0, %1, off"
                 :: "v"(ldsOff), "v"(gsrc) : "memory");
}
__device__ __forceinline__ void wait_async_le1() { asm volatile("s_wait_asynccnt 0x1" ::: "memory"); }
__device__ __forceinline__ void wait_async_le0() { asm volatile("s_wait_asynccnt 0x0" ::: "memory"); }

// ---------------------------------------------------------------------------
// Embedding + sinusoidal positional encoding.
// One block handles TOK consecutive tokens; weight rows (4KB each) are staged
// global->LDS with the async copy engine, double-buffered with 2 loads in
// flight. Each lane reads back only its own LDS slot, so no workgroup barrier
// is needed -- per-wave s_wait_asynccnt is the producer/consumer sync.
// Results stream out with non-temporal B128 stores (write-once data).
// ---------------------------------------------------------------------------
__global__ void __launch_bounds__(256)
emb_pos_kernel(const int* __restrict__ x, const float* __restrict__ w,
               float* __restrict__ out)
{
    __shared__ v4f smem[2][256];              // 8KB: double-buffered weight rows

    const int tid  = threadIdx.x;
    const int base = blockIdx.x * TOK;        // first token index of this block
    const int s0   = base % S;                // sequence position of token 0

    // low 32 bits of a generic pointer to LDS == workgroup-relative LDS offset
    const unsigned lds0 = (unsigned)(uintptr_t)(&smem[0][tid]);
    const unsigned lds1 = (unsigned)(uintptr_t)(&smem[1][tid]);

    const float kNegLn1e4OverD = -9.210340371976184f / (float)D;

    // prologue: issue load for token 0
    {
        const int tok = x[base];
        async_row_load16(lds0, (const v4f*)(w + (size_t)tok * D) + tid);
    }

#pragma unroll
    for (int k = 0; k < TOK; ++k) {
        if (k + 1 < TOK) {
            const int tnext = x[base + k + 1];
            async_row_load16(((k + 1) & 1) ? lds1 : lds0,
                             (const v4f*)(w + (size_t)tnext * D) + tid);
            wait_async_le1();                 // load k complete, k+1 in flight
        } else {
            wait_async_le0();                 // drain
        }

        // pos_table[s, d]: angle = s * 10000^(-(d & ~1)/1024); sin even, cos odd
        const float sf = (float)(s0 + k);
        const float a0 = sf * __expf((float)(4 * tid)     * kNegLn1e4OverD);
        const float a1 = sf * __expf((float)(4 * tid + 2) * kNegLn1e4OverD);

        v4f v = smem[k & 1][tid];
        v.x += __sinf(a0);
        v.y += __cosf(a0);
        v.z += __sinf(a1);
        v.w += __cosf(a1);

        __builtin_nontemporal_store(v, (v4f*)out + (size_t)(base + k) * (D / 4) + tid);
    }
}

// ---------------------------------------------------------------------------
// Mask build: mask[b,h,i,j] = (x[b,j]==PAD) || (i >= first[b])
// where first[b] = index of first pad (S-1 if none; faithful to the loop).
// Only two distinct row patterns exist per b: cache the pad row in LDS and
// stream 160KB of NT float4 stores per (b,h) block.
// ---------------------------------------------------------------------------
__global__ void __launch_bounds__(256)
mask_kernel(const int* __restrict__ x, float* __restrict__ mout)
{
    __shared__ v4f rowA[S / 4];               // pad row as floats (50 x float4)
    __shared__ int sFirst;

    const int b   = blockIdx.x;
    const int h   = blockIdx.y;
    const int tid = threadIdx.x;

    if (tid == 0) sFirst = S - 1;
    __syncthreads();

    float* rowAf = (float*)rowA;
    if (tid < S) {
        const bool pad = (x[b * S + tid] == PAD);
        rowAf[tid] = pad ? 1.0f : 0.0f;
        if (pad) atomicMin(&sFirst, tid);
    }
    __syncthreads();

    const int first = sFirst;
    v4f* base = (v4f*)(mout + ((size_t)b * H + h) * (size_t)(S * S));
    const v4f ones = (v4f){1.0f, 1.0f, 1.0f, 1.0f};

    // S rows x (S/4) float4 chunks = 10000 chunks per (b,h)
    for (int n = tid; n < S * (S / 4); n += 256) {
        const int i = n / (S / 4);
        const int c = n % (S / 4);
        const v4f val = (i >= first) ? ones : rowA[c];
        __builtin_nontemporal_store(val, base + (size_t)i * (S / 4) + c);
    }
}

// ---------------------------------------------------------------------------
extern "C" void kernel_launch(void* const* d_in, const int* in_sizes, int n_in,
                              void* d_out, int out_size, void* d_ws, size_t ws_size,
                              hipStream_t stream)
{
    const int*   x = (const int*)d_in[0];     // [B,S] token ids
    const float* w = (const float*)d_in[1];   // [32000, D]
    float* out  = (float*)d_out;              // [B,S,D] floats
    float* mout = out + (size_t)B * S * D;    // [B,H,S,S] floats (0/1)

    emb_pos_kernel<<<dim3(B * S / TOK), dim3(256), 0, stream>>>(x, w, out);
    mask_kernel<<<dim3(B, H), dim3(256), 0, stream>>>(x, mout);
}